// RaftGATLayer_26886495273044
// MI455X (gfx1250) — compile-verified
//
#include <hip/hip_runtime.h>

typedef float v2f __attribute__((ext_vector_type(2)));
typedef float v8f __attribute__((ext_vector_type(8)));
typedef int   v4i __attribute__((ext_vector_type(4)));

typedef __attribute__((address_space(1))) v4i* gvec_ptr;   // global int4*
typedef __attribute__((address_space(3))) v4i* lvec_ptr;   // LDS int4*

#define BATCH 8
#define NNODE 2048
#define DIM   128
#define NTILE (NNODE / 16)
#define NEG_SLOPE 0.2f

#if __has_builtin(__builtin_amdgcn_global_load_async_to_lds_b128) && \
    __has_builtin(__builtin_amdgcn_s_wait_asynccnt)
#define HAVE_ASYNC_LDS 1
#endif

__device__ __forceinline__ float leaky(float x) {
    return x > 0.0f ? x : NEG_SLOPE * x;
}

// ---------------------------------------------------------------------------
// Kernel 1: Wh = h @ W^T   (16384 x 128, K = 128) with fp32 WMMA 16x16x4
// ---------------------------------------------------------------------------
__global__ void __launch_bounds__(256) gat_wh_kernel(
    const float* __restrict__ h, const float* __restrict__ W,
    float* __restrict__ Wh)
{
    const int lane = threadIdx.x & 31;
    const int wave = threadIdx.x >> 5;
    const int tile = blockIdx.x * 8 + wave;          // 1024 tiles total
    const int rowbase = tile * 16;
    const int m  = lane & 15;
    const int cs = lane >> 4;                        // 0 or 1: K-pair selector

    v8f acc[8] = {};
    const float* hrow = h + (size_t)(rowbase + m) * DIM;

#pragma unroll 4
    for (int k0 = 0; k0 < DIM; k0 += 4) {
        const int k = k0 + 2 * cs;
        v2f a = *reinterpret_cast<const v2f*>(hrow + k);
#pragma unroll
        for (int nt = 0; nt < 8; ++nt) {
            const int o = nt * 16 + m;
            v2f b = *reinterpret_cast<const v2f*>(W + (size_t)o * DIM + k);
            acc[nt] = __builtin_amdgcn_wmma_f32_16x16x4_f32(
                false, a, false, b, (short)0, acc[nt], false, false);
        }
    }

    // D layout: acc[nt][v] = D[v + 8*cs][nt*16 + m]
#pragma unroll
    for (int nt = 0; nt < 8; ++nt) {
#pragma unroll
        for (int v = 0; v < 8; ++v) {
            const int r = rowbase + v + 8 * cs;
            Wh[(size_t)r * DIM + nt * 16 + m] = acc[nt][v];
        }
    }
}

// ---------------------------------------------------------------------------
// Kernel 2: s[n] = Wh[n] . a_src ; t[n] = Wh[n] . a_dst  (wave per node)
// ---------------------------------------------------------------------------
__global__ void __launch_bounds__(256) gat_st_kernel(
    const float* __restrict__ Wh, const float* __restrict__ a,
    float* __restrict__ s, float* __restrict__ t)
{
    const int lane = threadIdx.x & 31;
    const int wave = threadIdx.x >> 5;
    const int node = blockIdx.x * 8 + wave;          // 16384 nodes

    const float* wr = Wh + (size_t)node * DIM;
    float sv = 0.0f, tv = 0.0f;
#pragma unroll
    for (int q = 0; q < 4; ++q) {
        const int c = lane + 32 * q;
        const float w = wr[c];
        sv += w * a[c];
        tv += w * a[DIM + c];
    }
#pragma unroll
    for (int off = 16; off > 0; off >>= 1) {
        sv += __shfl_xor(sv, off, 32);
        tv += __shfl_xor(tv, off, 32);
    }
    if (lane == 0) {
        s[node] = sv;
        t[node] = tv;
    }
}

// ---------------------------------------------------------------------------
// Kernel 3: per-batch max of t  ->  T[b]
// ---------------------------------------------------------------------------
__global__ void __launch_bounds__(256) gat_tmax_kernel(
    const float* __restrict__ t, float* __restrict__ T)
{
    __shared__ float red[256];
    const int b = blockIdx.x;
    const int tid = threadIdx.x;
    float mx = -3.0e38f;
    for (int i = tid; i < NNODE; i += 256)
        mx = fmaxf(mx, t[(size_t)b * NNODE + i]);
    red[tid] = mx;
    __syncthreads();
    for (int sdim = 128; sdim > 0; sdim >>= 1) {
        if (tid < sdim) red[tid] = fmaxf(red[tid], red[tid + sdim]);
        __syncthreads();
    }
    if (tid == 0) T[b] = red[0];
}

// ---------------------------------------------------------------------------
// Kernel 4: fused masked-softmax attention + alpha @ Wh
// Block = 8 waves; each wave owns 16 rows.  The 16x128 Wh j-tile (one
// contiguous 8 KB block) is double-buffered in LDS via CDNA5 async
// global->LDS loads (ASYNCcnt), overlapping the next tile's fetch with
// the current tile's 32 fp32 WMMAs.
// ---------------------------------------------------------------------------
__global__ void __launch_bounds__(256) gat_attn_kernel(
    const int*   __restrict__ adj,
    const float* __restrict__ Wh,
    const float* __restrict__ s,
    const float* __restrict__ t,
    const float* __restrict__ T,
    float* __restrict__ out)
{
    __shared__ float lds_wh[2][16 * DIM];            // 2 x 8 KB double buffer

    const int tid  = threadIdx.x;
    const int lane = tid & 31;
    const int wave = tid >> 5;
    const int b     = blockIdx.x >> 4;               // 8 batches
    const int group = blockIdx.x & 15;               // 16 groups of 128 rows
    const int rowbase = group * 128 + wave * 16;     // within batch

    const int m  = lane & 15;
    const int cs = lane >> 4;

    const int   row  = rowbase + m;                  // this lane's i-row
    const float sv   = s[(size_t)b * NNODE + row];
    const float mrow = leaky(sv + T[b]);             // >= all e_ij (monotone)
    const int*   adjrow = adj + ((size_t)b * NNODE + row) * NNODE;
    const float* whbase = Wh + (size_t)b * NNODE * DIM;
    const float* tbase  = t + (size_t)b * NNODE;

    v8f acc[8] = {};
    float denom = 0.0f;

#ifdef HAVE_ASYNC_LDS
    // prologue: async-fetch tile 0 into buffer 0 (2 x b128 per lane)
#pragma unroll
    for (int r = 0; r < 2; ++r) {
        const int chunk = tid + 256 * r;             // 16-byte chunks
        __builtin_amdgcn_global_load_async_to_lds_b128(
            (gvec_ptr)(whbase + chunk * 4),
            (lvec_ptr)(&lds_wh[0][0] + chunk * 4),
            0, 0);
    }
#endif

    for (int jt = 0; jt < NTILE; ++jt) {
        const int j0  = jt * 16;
        const int cur = jt & 1;

#ifdef HAVE_ASYNC_LDS
        if (jt + 1 < NTILE) {
            // prefetch next tile into the other buffer (safe: previous
            // iteration ended with a barrier after computing on it)
            const float* src = whbase + (size_t)(j0 + 16) * DIM;
#pragma unroll
            for (int r = 0; r < 2; ++r) {
                const int chunk = tid + 256 * r;
                __builtin_amdgcn_global_load_async_to_lds_b128(
                    (gvec_ptr)(src + chunk * 4),
                    (lvec_ptr)(&lds_wh[cur ^ 1][0] + chunk * 4),
                    0, 0);
            }
            __builtin_amdgcn_s_wait_asynccnt(2);     // current tile landed
        } else {
            __builtin_amdgcn_s_wait_asynccnt(0);
        }
#else
        // fallback: synchronous staging (coalesced, 8 floats/thread)
#pragma unroll
        for (int r = 0; r < 8; ++r) {
            const int idx = r * 256 + tid;           // jr*128 + col
            lds_wh[cur][idx] = whbase[(size_t)(j0 + (idx >> 7)) * DIM
                                      + (idx & 127)];
        }
#endif
        __syncthreads();                             // all waves' tile ready

        // build probability A-fragments: lane covers cols {4q+2cs, 4q+2cs+1}
        v2f afrag[4];
#pragma unroll
        for (int q = 0; q < 4; ++q) {
            const int c = 4 * q + 2 * cs;
            const int2   ad = *reinterpret_cast<const int2*>(adjrow + j0 + c);
            const float2 tc = *reinterpret_cast<const float2*>(tbase + j0 + c);
            const float e0 = leaky(sv + tc.x) - mrow;
            const float e1 = leaky(sv + tc.y) - mrow;
            const float p0 = (ad.x != 0) ? __expf(e0) : 0.0f;
            const float p1 = (ad.y != 0) ? __expf(e1) : 0.0f;
            afrag[q] = (v2f){p0, p1};
            denom += p0 + p1;
        }

        // acc[nt] += P(16x16) @ Wh_tile(16x16)  via 4 chained K=4 WMMAs
        const float* whbuf = lds_wh[cur];
#pragma unroll
        for (int nt = 0; nt < 8; ++nt) {
            const int n = nt * 16 + m;
#pragma unroll
            for (int q = 0; q < 4; ++q) {
                const int k = 4 * q + 2 * cs;
                v2f bf = (v2f){whbuf[k * DIM + n], whbuf[(k + 1) * DIM + n]};
                acc[nt] = __builtin_amdgcn_wmma_f32_16x16x4_f32(
                    false, afrag[q], false, bf, (short)0, acc[nt],
                    false, false);
            }
        }
        __syncthreads();            // done with 'cur' before it is rewritten
    }

    // combine half-wave partial denominators: full denom for row m
    denom += __shfl_xor(denom, 16, 32);

    // fetch 1/denom for the rows this lane actually stores (v + 8*cs)
    float inv[8];
#pragma unroll
    for (int v = 0; v < 8; ++v) {
        const float dv = __shfl(denom, v + 8 * cs, 32);
        inv[v] = (dv > 0.0f) ? (1.0f / dv) : 0.0f;   // all-masked row -> 0
    }

#pragma unroll
    for (int nt = 0; nt < 8; ++nt) {
#pragma unroll
        for (int v = 0; v < 8; ++v) {
            const int r = rowbase + v + 8 * cs;
            out[((size_t)b * NNODE + r) * DIM + nt * 16 + m] =
                acc[nt][v] * inv[v];
        }
    }
}

// ---------------------------------------------------------------------------
extern "C" void kernel_launch(void* const* d_in, const int* in_sizes, int n_in,
                              void* d_out, int out_size, void* d_ws, size_t ws_size,
                              hipStream_t stream)
{
    const float* h   = (const float*)d_in[0];   // (8,2048,128)
    const int*   adj = (const int*)  d_in[1];   // (8,2048,2048)
    const float* W   = (const float*)d_in[2];   // (128,128)
    const float* a   = (const float*)d_in[3];   // (256,)
    float* out = (float*)d_out;                 // (8,2048,128)

    float* Wh = (float*)d_ws;                   // 16384*128 floats = 8 MB
    float* s  = Wh + (size_t)BATCH * NNODE * DIM;
    float* t  = s + (size_t)BATCH * NNODE;
    float* T  = t + (size_t)BATCH * NNODE;

    gat_wh_kernel<<<128, 256, 0, stream>>>(h, W, Wh);
    gat_st_kernel<<<2048, 256, 0, stream>>>(Wh, a, s, t);
    gat_tmax_kernel<<<BATCH, 256, 0, stream>>>(t, T);
    gat_attn_kernel<<<128, 256, 0, stream>>>(adj, Wh, s, t, T, out);
}